// PrototypeLoss_50208167690763
// MI455X (gfx1250) — compile-verified
//
#include <hip/hip_runtime.h>
#include <math.h>

// Problem constants (fixed by setup_inputs): B=262144, D=256, C=1000.
#define D_DIM      256
#define C_DIM      1000
#define C_PAD      1008          // pad classes to multiple of 16 for WMMA tiles
#define W_SLICE    16            // columns per scatter workgroup
#define N_SLICES   (D_DIM / W_SLICE)   // 16
#define NB_CHUNKS  32            // B-chunks per slice -> 512 WGs total
#define THREADS    256

// Workspace layout (floats):
//  [0, C_PAD*D_DIM)        : padded class sums  (1,032,192 B)
//  [SUMS_FLOATS]           : scalar accumulator for sum_c ||S_c||
//  [RN_OFF, RN_OFF+B)      : per-row inverse norms
#define SUMS_FLOATS (C_PAD * D_DIM)     // 258048
#define RN_OFF      (SUMS_FLOATS + 64)  // 258112 (256B-aligned)

typedef __attribute__((ext_vector_type(2))) float v2f;
typedef __attribute__((ext_vector_type(8))) float v8f;

// ---------------------------------------------------------------- zero init
__global__ void k_zero(float* __restrict__ p, int n) {
    int i = blockIdx.x * blockDim.x + threadIdx.x;
    if (i < n) p[i] = 0.0f;
}

// ------------------------------------------------- pass 1: per-row 1/||x||
// One wave (32 lanes) per row; each lane loads 2 x float4 (b128), wave reduce.
__global__ void __launch_bounds__(256)
k_rownorm(const float4* __restrict__ f4, float* __restrict__ rnorm, int B) {
    int wid  = blockIdx.x * (blockDim.x >> 5) + (threadIdx.x >> 5);
    int lane = threadIdx.x & 31;
    if (wid >= B) return;
    size_t base = (size_t)wid * (D_DIM / 4);
    float4 a = f4[base + lane];
    float4 b = f4[base + 32 + lane];
    float ss = a.x*a.x + a.y*a.y + a.z*a.z + a.w*a.w
             + b.x*b.x + b.y*b.y + b.z*b.z + b.w*b.w;
    #pragma unroll
    for (int m = 16; m > 0; m >>= 1) ss += __shfl_xor(ss, m, 32);
    if (lane == 0) {
        float n = sqrtf(ss);
        rnorm[wid] = 1.0f / fmaxf(n, 1e-12f);
    }
}

// --------------------------- pass 2: LDS-privatized segment-sum (scatter)
// WG (slice, chunk): owns columns [slice*16, slice*16+16) and rows
// [chunk*B/NB, ...). LDS table = 1000 x 16 fp32 (64,000 B). ds_add_f32
// accumulation, then global_atomic_add_f32 flush into the shared sums.
__global__ void __launch_bounds__(THREADS)
k_scatter(const float4* __restrict__ f4, const int* __restrict__ labels,
          const float* __restrict__ rnorm, float* __restrict__ sums, int B) {
    __shared__ float tab[C_DIM * W_SLICE]; // 64,000 B
    const int slice = blockIdx.x % N_SLICES;
    const int chunk = blockIdx.x / N_SLICES;
    const int tid   = threadIdx.x;

    for (int i = tid; i < C_DIM * W_SLICE; i += THREADS) tab[i] = 0.0f;
    __syncthreads();

    const int rows_per_chunk = B / NB_CHUNKS;      // 8192
    const int r0   = chunk * rows_per_chunk;
    const int sub  = tid & 3;                       // float4 within slice
    const int rofs = tid >> 2;                      // 0..63 rows / iter

    for (int it = 0; it < rows_per_chunk; it += 64) {
        int   row = r0 + it + rofs;
        int   lab = labels[row];
        float rn  = rnorm[row];
        float4 v  = f4[(size_t)row * (D_DIM / 4) + slice * (W_SLICE / 4) + sub];
        float* dst = &tab[lab * W_SLICE + sub * 4];
        atomicAdd(dst + 0, v.x * rn);
        atomicAdd(dst + 1, v.y * rn);
        atomicAdd(dst + 2, v.z * rn);
        atomicAdd(dst + 3, v.w * rn);
    }
    __syncthreads();

    for (int i = tid; i < C_DIM * W_SLICE; i += THREADS) {
        float v = tab[i];
        if (v != 0.0f) {
            int c = i >> 4, w = i & 15;
            unsafeAtomicAdd(&sums[c * D_DIM + slice * W_SLICE + w], v);
        }
    }
}

// ------------------- pass 3: sum_c ||S_c|| via V_WMMA_F32_16X16X4_F32
// One wave per 16-class tile. A = B = same 16x4 tile of S, so D accumulates
// the Gram matrix G = S_tile * S_tile^T over K=256 (64 chained WMMAs);
// diag(G)[m] = sum_k S[m][k]^2. Padded rows (1000..1007) are zero.
__global__ void __launch_bounds__(32)
k_classnorm(const float* __restrict__ sums, float* __restrict__ acc) {
    const int tile = blockIdx.x;          // 0..62
    const int lane = threadIdx.x;         // 0..31
    const int r    = tile * 16 + (lane & 15);
    const int kofs = (lane < 16) ? 0 : 2; // K pair held by this lane half
    const float* base = sums + (size_t)r * D_DIM + kofs;

    float part = 0.0f;
#if defined(__AMDGCN__) && __has_builtin(__builtin_amdgcn_wmma_f32_16x16x4_f32)
    v8f d = {0.f, 0.f, 0.f, 0.f, 0.f, 0.f, 0.f, 0.f};
    for (int kk = 0; kk < D_DIM; kk += 4) {
        v2f ab = *(const v2f*)(base + kk);   // lane: (S[r][kk+kofs], S[r][kk+kofs+1])
        d = __builtin_amdgcn_wmma_f32_16x16x4_f32(
                false, ab, false, ab, (short)0, d, false, false);
    }
    // Diagonal: m<8 -> (VGPR m, lane m); m>=8 -> (VGPR m-8, lane m+16).
    if (lane < 8)        part = sqrtf(fmaxf(d[lane], 0.0f));
    else if (lane >= 24) part = sqrtf(fmaxf(d[lane - 24], 0.0f));
#else
    if (lane < 16) {
        const float* rowp = sums + (size_t)(tile * 16 + lane) * D_DIM;
        float ss = 0.0f;
        for (int k = 0; k < D_DIM; ++k) ss += rowp[k] * rowp[k];
        part = sqrtf(ss);
    }
#endif
    #pragma unroll
    for (int m = 16; m > 0; m >>= 1) part += __shfl_xor(part, m, 32);
    if (lane == 0) unsafeAtomicAdd(acc, part);
}

// ------------------------------------------------------------- finalize
__global__ void k_final(const float* __restrict__ acc, float* __restrict__ out,
                        float invB) {
    out[0] = 1.0f - acc[0] * invB;
}

// ----------------------------------------------------------------- launch
extern "C" void kernel_launch(void* const* d_in, const int* in_sizes, int n_in,
                              void* d_out, int out_size, void* d_ws, size_t ws_size,
                              hipStream_t stream) {
    const float* features = (const float*)d_in[0];
    const int*   labels   = (const int*)d_in[1];
    (void)n_in; (void)out_size; (void)ws_size;   // C_DIM fixed at 1000

    float* ws    = (float*)d_ws;
    float* sums  = ws;                 // C_PAD x D_DIM
    float* acc   = ws + SUMS_FLOATS;   // scalar
    float* rnorm = ws + RN_OFF;        // B floats
    float* out   = (float*)d_out;

    const int B = in_sizes[0] / D_DIM; // 262144

    // zero padded sums + accumulator (every call: harness does not re-poison)
    int nz = SUMS_FLOATS + 1;
    k_zero<<<(nz + 255) / 256, 256, 0, stream>>>(ws, nz);

    // pass 1: inverse row norms (wave per row, 8 waves per block)
    k_rownorm<<<B / 8, 256, 0, stream>>>((const float4*)features, rnorm, B);

    // pass 2: column-sliced, LDS-privatized scatter-add
    k_scatter<<<N_SLICES * NB_CHUNKS, THREADS, 0, stream>>>(
        (const float4*)features, labels, rnorm, sums, B);

    // pass 3: sum of class-sum norms via WMMA Gram diagonals
    k_classnorm<<<C_PAD / 16, 32, 0, stream>>>(sums, acc);

    // finalize: loss = 1 - (sum_c ||S_c||) / B
    k_final<<<1, 1, 0, stream>>>(acc, out, 1.0f / (float)B);
}